// BinarizeLayer_36301063585944
// MI455X (gfx1250) — compile-verified
//
#include <hip/hip_runtime.h>

typedef __attribute__((ext_vector_type(2))) float v2f;
typedef __attribute__((ext_vector_type(4))) float v4f;
typedef __attribute__((ext_vector_type(8))) float v8f;

#define EPSV 0.001f
#define BIGC 1.0e4f
#define ROWS_PER_WAVE 16
#define WAVES_PER_BLOCK 8
#define ROWS_PER_BLOCK 128
#define XPAD 36   // 32 cols + 4 pad -> conflict-free LDS rows

__global__ void zero_loss_kernel(float* out, long long lossIdx) {
    out[lossIdx] = 0.0f;
}

// Per (feature,set): d[bin,row] via V_WMMA_F32_16X16X4_F32 with
//   A[m][:] = [center_m^2, center_m, 1, 0]   (m = bin)
//   B[:][n] = [1, -2*c_n, c_n^2, 0]          (n = row)
// Lane layout (wave32): lanes 0-15 hold K-slots {0,1}, lanes 16-31 hold {2,3};
// D: lane L holds bins r+8*(L/16) of row L%16 in 8 VGPRs.
__device__ __forceinline__ void soft_bin_set(float bx, float by,
                                             const float* centRow, int n, int hi,
                                             float& lossRow, int& gidx) {
    float center = centRow[n];
    v2f amat; amat.x = hi ? 1.0f : center * center; amat.y = hi ? 0.0f : center;
    v2f bmat; bmat.x = bx; bmat.y = by;
    v8f cz = {};
    v8f d = __builtin_amdgcn_wmma_f32_16x16x4_f32(false, amat, false, bmat,
                                                  (short)0, cz, false, false);
    // local min/argmin over this lane's 8 bins (ascending -> first occurrence)
    float dl = d[0]; int il = 0;
#pragma unroll
    for (int r = 1; r < 8; ++r) { if (d[r] < dl) { dl = d[r]; il = r; } }
    il += hi * 8;
    // branchless cross-half min + first-occurrence argmin
    float dmin = fminf(dl, __shfl_xor(dl, 16, 32));
    int cand = (dl == dmin) ? il : 64;           // loser half -> sentinel
    gidx = min(cand, __shfl_xor(cand, 16, 32));  // ties -> lower bin index
    // T1 = 1 softmax-weighted distance: sum(d*e)/sum(e), e = exp(dmin - d)
    float se = 0.0f, sde = 0.0f;
#pragma unroll
    for (int r = 0; r < 8; ++r) {
        float e = __expf(dmin - d[r]);
        se += e; sde += d[r] * e;
    }
    se  += __shfl_xor(se, 16, 32);
    sde += __shfl_xor(sde, 16, 32);
    lossRow = sde / se;
}

__global__ __launch_bounds__(256)
void binarize_kernel(const float* __restrict__ x,
                     const float* __restrict__ interval,
                     const float* __restrict__ interval2,
                     const float* __restrict__ i_min,
                     float* __restrict__ out,
                     float* __restrict__ lossOut,
                     int Btot) {
    __shared__ float xt[WAVES_PER_BLOCK][ROWS_PER_WAVE][XPAD];
    __shared__ float cent1[16][16];
    __shared__ float cent2[16][16];        // K2=8 padded to 16 with BIGC
    __shared__ unsigned char idx1s[ROWS_PER_BLOCK][16];
    __shared__ unsigned char idx2s[ROWS_PER_BLOCK][16];
    __shared__ float wsum[WAVES_PER_BLOCK];

    const int tid  = threadIdx.x;
    const int wave = tid >> 5;
    const int lane = tid & 31;
    const int n    = lane & 15;   // row within wave tile / A-row (bin) index
    const int hi   = lane >> 4;
    const int baseRow = blockIdx.x * ROWS_PER_BLOCK + wave * ROWS_PER_WAVE;

    // ---- centers: cumsum of enforce(interval) (forward: max(v, EPS)) ----
    {
        int f = tid >> 4, k = tid & 15;
        float s = i_min[f];
        for (int j = 0; j <= k; ++j) s += fmaxf(interval[f * 16 + j], EPSV);
        cent1[f][k] = s;
    }
    if (tid < 128) {
        int f = tid >> 3, k = tid & 7;
        float s = i_min[f];
        for (int j = 0; j <= k; ++j) s += fmaxf(interval2[f * 8 + j], EPSV);
        cent2[f][k] = s;
    } else {
        int u = tid - 128; int f = u >> 3, k = (u & 7) + 8;
        cent2[f][k] = BIGC;
    }

    // ---- stage this wave's 16 rows x 32 cols of x into LDS (float4) ----
#pragma unroll
    for (int i = 0; i < 4; ++i) {
        int idx = lane + 32 * i;          // 0..127 -> row(16) x col4(8)
        int row = idx >> 3;
        int c4  = idx & 7;
        int grow = baseRow + row;
        if (grow > Btot - 1) grow = Btot - 1;
        v4f val = *(const v4f*)(x + (size_t)grow * 32 + c4 * 4);
        *(v4f*)&xt[wave][row][c4 * 4] = val;
    }
    __syncthreads();

    const bool rowValid = (baseRow + n) < Btot;
    float lossAcc = 0.0f;

#pragma unroll 1
    for (int f = 0; f < 16; ++f) {
        float c  = xt[wave][n][16 + f];   // cont value of this lane's row
        float bx = hi ? c * c : 1.0f;
        float by = hi ? 0.0f : -2.0f * c;

        float lr1, lr2; int g1, g2;
        soft_bin_set(bx, by, &cent1[f][0], n, hi, lr1, g1);
        soft_bin_set(bx, by, &cent2[f][0], n, hi, lr2, g2);

        if (!hi) {
            idx1s[wave * 16 + n][f] = (unsigned char)g1;
            idx2s[wave * 16 + n][f] = (unsigned char)g2;
            if (rowValid) lossAcc += lr1 + lr2;
        }
    }
    __syncthreads();   // idx tables / x-tile visible across lanes

    // ---- write phase: 16 rows x 400 cols = 1600 float4 per wave ----
#pragma unroll 1
    for (int t = lane; t < 1600; t += 32) {
        int row = t / 100;
        int j   = t - row * 100;          // float4 column 0..99
        int grow = baseRow + row;
        if (grow >= Btot) continue;
        v4f v;
        if (j < 4) {                      // x_disc copy, cols 0..15
            int c0 = j * 4;
            v.x = xt[wave][row][c0 + 0]; v.y = xt[wave][row][c0 + 1];
            v.z = xt[wave][row][c0 + 2]; v.w = xt[wave][row][c0 + 3];
        } else if (j < 68) {              // out1 one-hot, cols 16..271
            int cc = j * 4 - 16;
            int f  = cc >> 4;
            int k0 = cc & 15;
            int id = idx1s[wave * 16 + row][f];
            v.x = (k0 + 0 == id) ? 1.0f : 0.0f;
            v.y = (k0 + 1 == id) ? 1.0f : 0.0f;
            v.z = (k0 + 2 == id) ? 1.0f : 0.0f;
            v.w = (k0 + 3 == id) ? 1.0f : 0.0f;
        } else {                          // out2 one-hot, cols 272..399
            int cc = j * 4 - 272;
            int f  = cc >> 3;
            int k0 = cc & 7;
            int id = idx2s[wave * 16 + row][f];
            v.x = (k0 + 0 == id) ? 1.0f : 0.0f;
            v.y = (k0 + 1 == id) ? 1.0f : 0.0f;
            v.z = (k0 + 2 == id) ? 1.0f : 0.0f;
            v.w = (k0 + 3 == id) ? 1.0f : 0.0f;
        }
        __builtin_nontemporal_store(v, (v4f*)(out + (size_t)grow * 400 + j * 4));
    }

    // ---- loss reduction: wave shfl -> LDS -> one atomic per block ----
#pragma unroll
    for (int off = 16; off > 0; off >>= 1) lossAcc += __shfl_xor(lossAcc, off, 32);
    if (lane == 0) wsum[wave] = lossAcc;
    __syncthreads();
    if (tid == 0) {
        float s = 0.0f;
#pragma unroll
        for (int i = 0; i < WAVES_PER_BLOCK; ++i) s += wsum[i];
        atomicAdd(lossOut, s * (1.0f / (float)Btot));
    }
}

extern "C" void kernel_launch(void* const* d_in, const int* in_sizes, int n_in,
                              void* d_out, int out_size, void* d_ws, size_t ws_size,
                              hipStream_t stream) {
    const float* x         = (const float*)d_in[0];
    const float* interval  = (const float*)d_in[1];
    const float* interval2 = (const float*)d_in[2];
    const float* i_min     = (const float*)d_in[3];
    float* out = (float*)d_out;
    int Btot = in_sizes[0] / 32;
    long long lossIdx = (long long)out_size - 1;

    zero_loss_kernel<<<1, 1, 0, stream>>>(out, lossIdx);
    int grid = (Btot + ROWS_PER_BLOCK - 1) / ROWS_PER_BLOCK;
    binarize_kernel<<<grid, 256, 0, stream>>>(x, interval, interval2, i_min,
                                              out, out + lossIdx, Btot);
}